// LightGCN_50302656971174
// MI455X (gfx1250) — compile-verified
//
#include <hip/hip_runtime.h>
#include <hip/hip_bf16.h>

#define EMBED 64
#define NUM_LAYERS 3
#define EDGE_CHUNK 64   // contiguous edges per 16-lane group (register row-accumulation)

typedef __attribute__((ext_vector_type(2))) float v2f;
typedef __attribute__((ext_vector_type(8))) float v8f;

// ---------------- init: feats = concat(uEmbd, iEmbd); acc = feats ----------------
__global__ void lgcn_init(const float4* __restrict__ uE, const float4* __restrict__ iE,
                          float4* __restrict__ f, float4* __restrict__ acc,
                          int uQuads, int totQuads) {
  for (int idx = blockIdx.x * blockDim.x + threadIdx.x; idx < totQuads;
       idx += gridDim.x * blockDim.x) {
    float4 v = (idx < uQuads) ? uE[idx] : iE[idx - uQuads];
    f[idx] = v;
    acc[idx] = v;
  }
}

// ---------------- zero a buffer ----------------
__global__ void lgcn_zero(float4* __restrict__ dst, int totQuads) {
  float4 z = make_float4(0.f, 0.f, 0.f, 0.f);
  for (int idx = blockIdx.x * blockDim.x + threadIdx.x; idx < totQuads;
       idx += gridDim.x * blockDim.x) {
    dst[idx] = z;
  }
}

// ---------------- acc += layer output ----------------
__global__ void lgcn_acc_add(float4* __restrict__ acc, const float4* __restrict__ src,
                             int totQuads) {
  for (int idx = blockIdx.x * blockDim.x + threadIdx.x; idx < totQuads;
       idx += gridDim.x * blockDim.x) {
    float4 a = acc[idx];
    float4 s = src[idx];
    a.x += s.x; a.y += s.y; a.z += s.z; a.w += s.w;
    acc[idx] = a;
  }
}

// ---------------- forced-native f32 atomic add (no-return; STOREcnt-tracked) ----------------
__device__ __forceinline__ void atomic_add_f32_native(float* p, float v) {
#if defined(__gfx1250__)
  asm volatile("global_atomic_add_f32 %0, %1, off" : : "v"(p), "v"(v) : "memory");
#else
  __hip_atomic_fetch_add(p, v, __ATOMIC_RELAXED, __HIP_MEMORY_SCOPE_AGENT);
#endif
}

// ---------------- edge-parallel SpMM with register row-accumulation ----------------
// dst[rows[e]] += vals[e] * src[cols[e]], rows/cols over [N], src/dst are [N,64] f32.
// 16 lanes per edge group, float4 (4 dims) per lane. Each group owns EDGE_CHUNK
// *contiguous* edges: the user-half of the edge list is sorted by row (np.unique),
// so consecutive edges usually share a row -> accumulate in a float4 register and
// flush 4 atomics only on row change / chunk end (~2x fewer atomics overall).
__global__ void lgcn_spmm(const int* __restrict__ rows, const int* __restrict__ cols,
                          const float* __restrict__ vals,
                          const float* __restrict__ src, float* __restrict__ dst,
                          int nnz) {
  int t = blockIdx.x * blockDim.x + threadIdx.x;
  int lane = t & 15;                 // 4 consecutive dims per lane
  int group = t >> 4;
  int eBeg = group * EDGE_CHUNK;
  if (eBeg >= nnz) return;
  int eEnd = eBeg + EDGE_CHUNK;
  if (eEnd > nnz) eEnd = nnz;

  int curRow = rows[eBeg];
  float ax = 0.f, ay = 0.f, az = 0.f, aw = 0.f;

  for (int e = eBeg; e < eEnd; ++e) {
    int r = rows[e];
    int c = cols[e];
    float v = vals[e];
    if (r != curRow) {               // uniform within the 16-lane group
      float* dp = dst + (size_t)curRow * EMBED + lane * 4;
      atomic_add_f32_native(dp + 0, ax);
      atomic_add_f32_native(dp + 1, ay);
      atomic_add_f32_native(dp + 2, az);
      atomic_add_f32_native(dp + 3, aw);
      ax = ay = az = aw = 0.f;
      curRow = r;
    }
    const float4* sp = (const float4*)(src + (size_t)c * EMBED) + lane;
    float4 x = *sp;                  // global_load_b128, L2-resident gather
    ax += v * x.x; ay += v * x.y; az += v * x.z; aw += v * x.w;
  }
  float* dp = dst + (size_t)curRow * EMBED + lane * 4;
  atomic_add_f32_native(dp + 0, ax);
  atomic_add_f32_native(dp + 1, ay);
  atomic_add_f32_native(dp + 2, az);
  atomic_add_f32_native(dp + 3, aw);
}

// ---------------- final gather + dot via V_WMMA_F32_16X16X4_F32 ----------------
// One wave handles 16 batch pairs. A = 16 user rows x 4 dims, B = 4 dims x 16 item
// cols; chain 16 WMMAs over K=64. D[m][m] = dot(ue[m], ie[m]). Diagonal extraction
// per ISA 16x16 f32 C/D layout: (m,m) sits in lane m, VGPR m for m<8 and in
// lane m+16, VGPR m-8 for m>=8.
__global__ void lgcn_dot_wmma(const float* __restrict__ acc,
                              const int* __restrict__ uIdx, const int* __restrict__ iIdx,
                              float* __restrict__ out, int nWaves, int U_num) {
  int wave = (blockIdx.x * blockDim.x + threadIdx.x) >> 5;
  if (wave >= nWaves) return;              // wave-uniform: EXEC stays all-1s
  int lane = threadIdx.x & 31;
  int m = lane & 15;                       // row (user) / col (item) = batch slot
  int koff = (lane >> 4) << 1;             // lanes 0-15: K pair {0,1}; lanes 16-31: {2,3}
  int b = wave * 16 + m;
  size_t urow = (size_t)uIdx[b] * EMBED;
  size_t irow = ((size_t)iIdx[b] + (size_t)U_num) * EMBED;

  v8f c = {};
#pragma unroll
  for (int k0 = 0; k0 < EMBED; k0 += 4) {
    const float* ua = acc + urow + k0 + koff;
    const float* ia = acc + irow + k0 + koff;
    v2f a;  a.x  = ua[0]; a.y  = ua[1];
    v2f bb; bb.x = ia[0]; bb.y = ia[1];
    c = __builtin_amdgcn_wmma_f32_16x16x4_f32(
        /*neg_a=*/false, a, /*neg_b=*/false, bb,
        /*c_mod=*/(short)0, c, /*reuse_a=*/false, /*reuse_b=*/false);
  }

  // pick diagonal element for this lane (if it owns one)
  int dm = -1;
  if (lane < 8) dm = lane;                 // D[m][m], m = 0..7
  else if (lane >= 24) dm = lane - 16;     // D[m][m], m = 8..15
  if (dm >= 0) {
    int idx = dm & 7;
    float d = c[0];
    if (idx == 1) d = c[1];
    if (idx == 2) d = c[2];
    if (idx == 3) d = c[3];
    if (idx == 4) d = c[4];
    if (idx == 5) d = c[5];
    if (idx == 6) d = c[6];
    if (idx == 7) d = c[7];
    // final = acc / (L+1); dot scales by 1/16
    out[wave * 16 + dm] = d * 0.0625f;
  }
}

// ---------------- scalar tail (batch % 16 != 0; unused at B=8192) ----------------
__global__ void lgcn_dot_scalar(const float* __restrict__ acc,
                                const int* __restrict__ uIdx, const int* __restrict__ iIdx,
                                float* __restrict__ out, int start, int batch, int U_num) {
  int b = start + blockIdx.x * blockDim.x + threadIdx.x;
  if (b >= batch) return;
  const float* pu = acc + (size_t)uIdx[b] * EMBED;
  const float* pi = acc + ((size_t)iIdx[b] + (size_t)U_num) * EMBED;
  float s = 0.f;
#pragma unroll
  for (int k = 0; k < EMBED; ++k) s += pu[k] * pi[k];
  out[b] = s * 0.0625f;
}

extern "C" void kernel_launch(void* const* d_in, const int* in_sizes, int n_in,
                              void* d_out, int out_size, void* d_ws, size_t ws_size,
                              hipStream_t stream) {
  const float* uE  = (const float*)d_in[0];
  const float* iE  = (const float*)d_in[1];
  const int* rows  = (const int*)d_in[2];
  const int* cols  = (const int*)d_in[3];
  const float* vals = (const float*)d_in[4];
  const int* uIdx  = (const int*)d_in[5];
  const int* iIdx  = (const int*)d_in[6];

  int U = in_sizes[0] / EMBED;
  int I = in_sizes[1] / EMBED;
  int N = U + I;
  int nnz = in_sizes[2];
  int batch = in_sizes[5];

  // workspace: f | g | acc  (each N*EMBED floats; 3 x 38.4 MB at reference sizes)
  float* f   = (float*)d_ws;
  float* g   = f + (size_t)N * EMBED;
  float* acc = g + (size_t)N * EMBED;

  int totQuads = N * (EMBED / 4);
  int uQuads = U * (EMBED / 4);

  const int EB = 256;
  int eGrid = 2048;  // grid-stride over 2.4M float4s

  lgcn_init<<<eGrid, EB, 0, stream>>>((const float4*)uE, (const float4*)iE,
                                      (float4*)f, (float4*)acc, uQuads, totQuads);

  // 16 groups per 256-thread block, EDGE_CHUNK contiguous edges per group
  long long groups = ((long long)nnz + EDGE_CHUNK - 1) / EDGE_CHUNK;
  long long wantBlocks = (groups + 15) / 16;
  int sGrid = (int)(wantBlocks > (1 << 20) ? (1 << 20)
                                           : (wantBlocks < 1 ? 1 : wantBlocks));

  float* src = f;
  float* dst = g;
  for (int l = 0; l < NUM_LAYERS; ++l) {
    lgcn_zero<<<eGrid, EB, 0, stream>>>((float4*)dst, totQuads);
    lgcn_spmm<<<sGrid, EB, 0, stream>>>(rows, cols, vals, src, dst, nnz);
    lgcn_acc_add<<<eGrid, EB, 0, stream>>>((float4*)acc, (const float4*)dst, totQuads);
    float* t = src; src = dst; dst = t;
  }

  int nWaves = batch / 16;
  if (nWaves > 0) {
    int threads = nWaves * 32;
    lgcn_dot_wmma<<<(threads + 255) / 256, 256, 0, stream>>>(acc, uIdx, iIdx,
                                                             (float*)d_out, nWaves, U);
  }
  int tail = batch - nWaves * 16;
  if (tail > 0) {
    lgcn_dot_scalar<<<(tail + 63) / 64, 64, 0, stream>>>(acc, uIdx, iIdx,
                                                         (float*)d_out, nWaves * 16,
                                                         batch, U);
  }
}